// HierarchicalSoftmaxHead_32298154066259
// MI455X (gfx1250) — compile-verified
//
#include <hip/hip_runtime.h>
#include <hip/hip_bf16.h>
#include <math.h>

// HierarchicalSoftmaxHead, fused gather+matvec via CDNA5 f32 WMMA.
//
// Reference does h[T,H] @ W.T[H,N] then gathers 16 path logits/token.
// We compute ONLY the 16 needed dots per token:
//   logit[t][j] = dot(h[t], W[path_nodes[target[t]][j]])
// One wave32 per token, V_WMMA_F32_16X16X4_F32:
//   A (16x4, M=step, K=h-chunk) = gathered weight rows
//   B (4x16, K=h-chunk, N=any)  = hidden chunk broadcast across columns
//   C (16x16 f32)               = logits replicated across columns; read col 0.
// 32-bit A/B VGPR layout (ISA 7.12.2): lanes 0-15 hold K={0,1} in v0/v1,
// lanes 16-31 hold K={2,3}; lane%16 = M (A) / N (B).

typedef __attribute__((ext_vector_type(2))) float v2f;
typedef __attribute__((ext_vector_type(8))) float v8f;

#define HSM_L 16  // max path length (reference constant)

__global__ __launch_bounds__(256) void hsm_path_bce_kernel(
    const float* __restrict__ hidden,        // [T, H]
    const int*   __restrict__ target_ids,    // [T]
    const float* __restrict__ all_weights,   // [N, H]
    const int*   __restrict__ path_nodes,    // [V, L]
    const float* __restrict__ path_targets,  // [V, L]
    const float* __restrict__ path_masks,    // [V, L]
    float2*      __restrict__ partials,      // [T] (bce_sum, mask_sum)
    int T, int H, int V)
{
    const int lane  = threadIdx.x & 31;
    const int token = blockIdx.x * (blockDim.x >> 5) + (threadIdx.x >> 5);
    if (token >= T) return;            // wave-uniform (grid is exact anyway)

    const int m = lane & 15;           // path step handled by this lane pair
    int tid = target_ids[token];
    tid = tid < 0 ? 0 : (tid > V - 1 ? V - 1 : tid);

    const int   pidx = tid * HSM_L + m;
    const int   node = path_nodes[pidx];
    const float tgt  = path_targets[pidx];
    const float msk  = (lane < 16) ? path_masks[pidx] : 0.0f;
    const int   widx = node < 0 ? 0 : node;     // clip padding (-1) like reference

    const float* __restrict__ wrow = all_weights + (size_t)widx  * (size_t)H;
    const float* __restrict__ hrow = hidden      + (size_t)token * (size_t)H;
    const int kk = (lane < 16) ? 0 : 2;         // K sub-block for this half-wave

    v8f c0 = {};
    v8f c1 = {};
    // 1024 / 8 = 128 iterations, 2 independent WMMA accumulator chains.
    for (int k0 = 0; k0 < H; k0 += 8) {
        v2f a0 = *(const v2f*)(wrow + k0 + kk);
        v2f b0 = *(const v2f*)(hrow + k0 + kk);
        v2f a1 = *(const v2f*)(wrow + k0 + 4 + kk);
        v2f b1 = *(const v2f*)(hrow + k0 + 4 + kk);
        // 8 args: (neg_a, A, neg_b, B, c_mod, C, reuse_a, reuse_b)
        c0 = __builtin_amdgcn_wmma_f32_16x16x4_f32(false, a0, false, b0,
                                                   (short)0, c0, false, false);
        c1 = __builtin_amdgcn_wmma_f32_16x16x4_f32(false, a1, false, b1,
                                                   (short)0, c1, false, false);
    }
    v8f c = c0 + c1;

    // Extract logits: C layout has row M=g in c[g] (lanes 0-15) and row M=g+8
    // in c[g] (lanes 16-31). Column 0 lives at lane 0 / lane 16.
    // Lane j (j = m) wants logit of step M = j.
    float logit = 0.0f;
#pragma unroll
    for (int g = 0; g < 8; ++g) {
        float lo = __shfl(c[g], 0, 32);    // out[M=g  ][N=0]
        float hi = __shfl(c[g], 16, 32);   // out[M=g+8][N=0]
        if ((m & 7) == g) logit = ((m & 8) == 0) ? lo : hi;
    }

    // Stable BCE-with-logits: max(x,0) - x*t + log1p(exp(-|x|))
    const float absl = fabsf(logit);
    const float bce  = fmaxf(logit, 0.0f) - logit * tgt + log1pf(expf(-absl));

    float s  = bce * msk;   // lanes >= 16 carry msk = 0
    float cn = msk;
#pragma unroll
    for (int off = 16; off > 0; off >>= 1) {
        s  += __shfl_xor(s,  off, 32);
        cn += __shfl_xor(cn, off, 32);
    }
    if (lane == 0) partials[token] = make_float2(s, cn);
}

__global__ __launch_bounds__(256) void hsm_reduce_kernel(
    const float2* __restrict__ partials, int T, float* __restrict__ out)
{
    __shared__ float ssum[256];
    __shared__ float scnt[256];
    float s = 0.0f, cn = 0.0f;
    for (int i = threadIdx.x; i < T; i += 256) {
        float2 p = partials[i];
        s  += p.x;
        cn += p.y;
    }
    ssum[threadIdx.x] = s;
    scnt[threadIdx.x] = cn;
    __syncthreads();
    for (int off = 128; off > 0; off >>= 1) {
        if ((int)threadIdx.x < off) {
            ssum[threadIdx.x] += ssum[threadIdx.x + off];
            scnt[threadIdx.x] += scnt[threadIdx.x + off];
        }
        __syncthreads();
    }
    if (threadIdx.x == 0)
        out[0] = (scnt[0] > 0.0f) ? (ssum[0] / scnt[0]) : 0.0f;
}

extern "C" void kernel_launch(void* const* d_in, const int* in_sizes, int n_in,
                              void* d_out, int out_size, void* d_ws, size_t ws_size,
                              hipStream_t stream) {
    const float* hidden       = (const float*)d_in[0];  // [B,S,H] f32
    const int*   target_ids   = (const int*)  d_in[1];  // [B,S]   i32
    const float* all_weights  = (const float*)d_in[2];  // [N,H]   f32
    const int*   path_nodes   = (const int*)  d_in[3];  // [V,L]   i32
    const float* path_targets = (const float*)d_in[4];  // [V,L]   f32
    const float* path_masks   = (const float*)d_in[5];  // [V,L]   f32
    float*       out          = (float*)d_out;

    const int T = in_sizes[1];              // B*S tokens
    const int H = in_sizes[0] / T;          // hidden size (1024)
    const int V = in_sizes[3] / HSM_L;      // vocab size

    float2* partials = (float2*)d_ws;       // T * 8 bytes of scratch

    const int waves_per_block = 8;          // 256 threads = 8 wave32
    const int blocks = (T + waves_per_block - 1) / waves_per_block;

    hsm_path_bce_kernel<<<blocks, 256, 0, stream>>>(
        hidden, target_ids, all_weights, path_nodes, path_targets, path_masks,
        partials, T, H, V);

    hsm_reduce_kernel<<<1, 256, 0, stream>>>(partials, T, out);
}